// GHMC_loss_6416681140472
// MI455X (gfx1250) — compile-verified
//
#include <hip/hip_runtime.h>
#include <hip/hip_bf16.h>

// GHM-C loss, single pass:
//   loss = (1/n) * sum_b S_b / max(C_b, 1),  n = max(#nonempty bins, 1)
// where C_b = valid count in bin b, S_b = valid-masked BCE sum in bin b,
// bin = min(floor(10 * |sigmoid(p0-p1) - onehot0|), 9)  (shared by both classes).

#define GHM_BINS 10
#define HSTRIDE  21   // odd stride -> 32 lanes map to 32 distinct LDS banks
#define BS       256

typedef __attribute__((ext_vector_type(2))) float v2f;
typedef __attribute__((ext_vector_type(8))) float v8f;

__device__ __forceinline__ void ghm_row(float p0, float p1, int t,
                                        float w0, float w1,
                                        float* mycnt, float* mysum) {
    const bool  tz = (t == 0);
    // softmax over 2 classes: s0 = sigmoid(p0 - p1)
    const float s0 = 1.0f / (1.0f + __expf(p1 - p0));
    const float g  = fabsf(s0 - (tz ? 1.0f : 0.0f));
    const int   bin = (int)fminf(g * 10.0f, 9.0f);
    const float v0 = (w0 > 0.0f) ? 1.0f : 0.0f;
    const float v1 = (w1 > 0.0f) ? 1.0f : 0.0f;
    const float cw = v0 + v1;
    if (cw > 0.0f) {
        // bce_c = max(p,0) - p*t_c + log1p(exp(-|p|))
        const float b0 = fmaxf(p0, 0.0f) - (tz ? p0 : 0.0f)
                       + __logf(1.0f + __expf(-fabsf(p0)));
        const float b1 = fmaxf(p1, 0.0f) - (tz ? 0.0f : p1)
                       + __logf(1.0f + __expf(-fabsf(p1)));
        const float bw = v0 * b0 + v1 * b1;
        // per-lane private slots -> ds_add_f32, conflict-free, deterministic
        atomicAdd(&mycnt[bin], cw);
        atomicAdd(&mysum[bin], bw);
    }
}

__global__ void __launch_bounds__(BS)
ghm_main_kernel(const float* __restrict__ pred,
                const int*   __restrict__ target,
                const float* __restrict__ lw,
                float* __restrict__ ws, int Brows) {
    __shared__ float hcnt[BS * HSTRIDE];
    __shared__ float hsum[BS * HSTRIDE];
    const int tid = threadIdx.x;
    for (int i = tid; i < BS * HSTRIDE; i += BS) { hcnt[i] = 0.0f; hsum[i] = 0.0f; }
    __syncthreads();

    float* mycnt = &hcnt[tid * HSTRIDE];
    float* mysum = &hsum[tid * HSTRIDE];

    const int P   = Brows >> 1;                 // row pairs
    const int gid = (int)blockIdx.x * BS + tid;
    const int T   = (int)gridDim.x * BS;

    const float4* pred4 = (const float4*)pred;  // 2 rows of pred per float4
    const float4* lw4   = (const float4*)lw;    // 2 rows of label_weight
    const int2*   tgt2  = (const int2*)target;  // 2 targets

    for (int idx = gid; idx < P; idx += T) {
        float4 p  = pred4[idx];
        float4 w  = lw4[idx];
        int2   tg = tgt2[idx];
        ghm_row(p.x, p.y, tg.x, w.x, w.y, mycnt, mysum);
        ghm_row(p.z, p.w, tg.y, w.z, w.w, mycnt, mysum);
    }
    if ((Brows & 1) && gid == 0) {
        const int r = Brows - 1;
        ghm_row(pred[2 * r], pred[2 * r + 1], target[r],
                lw[2 * r], lw[2 * r + 1], mycnt, mysum);
    }
    __syncthreads();

    // Block-level combine: counts -> cols 0..9, bce sums -> cols 16..25,
    // pad cols written as 0 so every ws row is fully defined each call.
    if (tid < 32) {
        float v = 0.0f;
        if (tid < GHM_BINS) {
            for (int k = 0; k < BS; ++k) v += hcnt[k * HSTRIDE + tid];
        } else if (tid >= 16 && tid < 16 + GHM_BINS) {
            const int b = tid - 16;
            for (int k = 0; k < BS; ++k) v += hsum[k * HSTRIDE + b];
        }
        ws[(int)blockIdx.x * 32 + tid] = v;
    }
}

// Cross-block reduction of nb x 32 partials via V_WMMA_F32_16X16X4_F32:
// A = ones(16x4)  =>  D[m][n] = sum_k B[k][n]  (column sums), accumulated in C.
__global__ void __launch_bounds__(32)
ghm_finalize_kernel(const float* __restrict__ ws, float* __restrict__ out, int nb) {
    const int lane = threadIdx.x;        // one full wave32, EXEC all ones
    const int half = lane >> 4;          // 0: K rows r, r+1 | 1: rows r+2, r+3
    const int col  = lane & 15;

    v2f a; a.x = 1.0f; a.y = 1.0f;       // all-ones A (16x4 f32)
    v8f acc0 = {0.f, 0.f, 0.f, 0.f, 0.f, 0.f, 0.f, 0.f};  // counts tile
    v8f acc1 = {0.f, 0.f, 0.f, 0.f, 0.f, 0.f, 0.f, 0.f};  // bce-sum tile

    for (int r = 0; r < nb; r += 4) {
        const int row = r + half * 2;
        v2f b0, b1;
        b0.x = ws[(row + 0) * 32 + col];
        b0.y = ws[(row + 1) * 32 + col];
        b1.x = ws[(row + 0) * 32 + 16 + col];
        b1.y = ws[(row + 1) * 32 + 16 + col];
        acc0 = __builtin_amdgcn_wmma_f32_16x16x4_f32(false, a, false, b0,
                                                     (short)0, acc0, false, false);
        acc1 = __builtin_amdgcn_wmma_f32_16x16x4_f32(false, a, false, b1,
                                                     (short)0, acc1, false, false);
    }

    // D VGPR0, lanes 0..15 = row M=0 = column sums
    __shared__ float cbins[16];
    __shared__ float sbins[16];
    if (lane < 16) { cbins[lane] = acc0[0]; sbins[lane] = acc1[0]; }
    __syncthreads();

    if (lane == 0) {
        int   n   = 0;
        float acc = 0.0f;
        for (int b = 0; b < GHM_BINS; ++b) {
            const float cb = cbins[b];
            if (cb > 0.0f) { ++n; acc += sbins[b] / fmaxf(cb, 1.0f); }
        }
        if (n < 1) n = 1;
        out[0] = acc / (float)n;
    }
}

extern "C" void kernel_launch(void* const* d_in, const int* in_sizes, int n_in,
                              void* d_out, int out_size, void* d_ws, size_t ws_size,
                              hipStream_t stream) {
    const float* pred   = (const float*)d_in[0];
    const int*   target = (const int*)d_in[1];
    const float* lw     = (const float*)d_in[2];
    const int    Brows  = in_sizes[1];   // target has B elements

    // nb partial rows of 32 floats in workspace; power of two >= 4.
    int nb = 1024;
    while ((size_t)nb * 32 * sizeof(float) > ws_size && nb > 4) nb >>= 1;

    ghm_main_kernel<<<nb, BS, 0, stream>>>(pred, target, lw, (float*)d_ws, Brows);
    ghm_finalize_kernel<<<1, 32, 0, stream>>>((const float*)d_ws, (float*)d_out, nb);

    (void)n_in; (void)out_size;
}